// RungeKuttaIntegratorCell_78769700208898
// MI455X (gfx1250) — compile-verified
//
#include <hip/hip_runtime.h>

typedef float v2f __attribute__((ext_vector_type(2)));
typedef float v8f __attribute__((ext_vector_type(8)));

#define T_STEPS 2000000
#define CHUNKS  32768
#define CLEN    64
#define NBLK    (CLEN / 16)

// workspace layout (float offsets)
#define WS_ML 0                       // M^CLEN (16)
#define WS_P  16                      // Kogge-Stone powers ML^(32*2^k), k=0..9 (160)
#define WS_A  176                     // WMMA A tables, lane-swizzled (3*9*2*32 = 1728)
#define WS_R  (WS_A + 1728)           // chunk aggregates r_c (CHUNKS*4)
#define WS_XS (WS_R + CHUNKS * 4)     // chunk start states (CHUNKS*4)

// ---------------- K0: build all constant matrices / tables (1 thread) ----------------
__device__ inline void mm44(const float* a, const float* b, float* c) {
  for (int i = 0; i < 4; i++)
    for (int j = 0; j < 4; j++) {
      float s = 0.f;
      for (int k = 0; k < 4; k++) s += a[i * 4 + k] * b[k * 4 + j];
      c[i * 4 + j] = s;
    }
}
__device__ inline void cp16(float* d, const float* s) { for (int i = 0; i < 16; i++) d[i] = s[i]; }

__global__ void k0_setup(const float* __restrict__ cD, const float* __restrict__ mD,
                         const float* __restrict__ kD, const float* __restrict__ dtD,
                         float* __restrict__ ws) {
  if (threadIdx.x != 0 || blockIdx.x != 0) return;
  const float c1 = cD[0], c2 = cD[1], c3 = cD[2];
  const float m1 = mD[0], m2 = mD[1];
  const float k1 = kD[0], k2 = kD[1], k3 = kD[2];
  const float dt = dtD[0];

  float A[16] = {
      0.f, 1.f, 0.f, 0.f,
      -(k1 + k2) / m1, -(c1 + c2) / m1, k2 / m1, c2 / m1,
      0.f, 0.f, 0.f, 1.f,
      k2 / m2, c2 / m2, -(k3 + k2) / m2, -(c3 + c2) / m2};
  float dA[16];
  for (int i = 0; i < 16; i++) dA[i] = dt * A[i];
  const float I[16] = {1, 0, 0, 0, 0, 1, 0, 0, 0, 0, 1, 0, 0, 0, 0, 1};

  // M = sum_{i=0..4} (dtA)^i / i!
  float M[16], Pw[16], tmp[16];
  cp16(M, I); cp16(Pw, I);
  for (int i = 1; i <= 4; i++) {
    mm44(Pw, dA, tmp);
    for (int j = 0; j < 16; j++) Pw[j] = tmp[j] / (float)i;
    for (int j = 0; j < 16; j++) M[j] += Pw[j];
  }
  // Nd = dt * (sum_{i=0..3} (dtA)^i/(i+1)!) * B,   B = col0: e1/m1, col1: e3/m2
  float S[16], Q[16];
  cp16(S, I); cp16(Q, I);
  for (int i = 1; i <= 3; i++) {
    mm44(Q, dA, tmp);
    for (int j = 0; j < 16; j++) Q[j] = tmp[j] / (float)(i + 1);
    for (int j = 0; j < 16; j++) S[j] += Q[j];
  }
  float Nd[8];  // 4x2
  for (int r = 0; r < 4; r++) {
    Nd[r * 2 + 0] = dt * S[r * 4 + 1] / m1;
    Nd[r * 2 + 1] = dt * S[r * 4 + 3] / m2;
  }

  // powers M^0..M^16
  float Mp[17][16];
  cp16(Mp[0], I);
  for (int t = 1; t <= 16; t++) mm44(Mp[t - 1], M, Mp[t]);

  // MN[x] = M^x * Nd (4x2), x=0..15
  float MN[16][8];
  for (int x = 0; x < 16; x++)
    for (int r = 0; r < 4; r++)
      for (int cc = 0; cc < 2; cc++) {
        float s = 0.f;
        for (int k = 0; k < 4; k++) s += Mp[x][r * 4 + k] * Nd[k * 2 + cc];
        MN[x][r * 2 + cc] = s;
      }

  // ML = M^64
  float ML[16];
  mm44(Mp[16], Mp[16], tmp);  // M^32
  mm44(tmp, tmp, ML);         // M^64
  for (int i = 0; i < 16; i++) ws[WS_ML + i] = ML[i];

  // Kogge-Stone powers: P_k = ML^(32 * 2^k), k = 0..9
  float X[16];
  cp16(X, ML);
  for (int i = 0; i < 5; i++) { mm44(X, X, tmp); cp16(X, tmp); }  // ML^32
  for (int k = 0; k < 10; k++) {
    for (int i = 0; i < 16; i++) ws[WS_P + k * 16 + i] = X[i];
    mm44(X, X, tmp); cp16(X, tmp);
  }

  // WMMA A tables: out 0 = z1 rows, out 1 = z2 rows, out 2 = carry (rows 0..3)
  // slice 0 multiplies x_base; slices 1..8 multiply (u_{2(s-1)}, u_{2(s-1)+1})
  float At[3][9][16][4] = {};
  for (int t = 0; t < 16; t++) {
    for (int k = 0; k < 4; k++) {
      At[0][0][t][k] = Mp[t + 1][0 * 4 + k];
      At[1][0][t][k] = Mp[t + 1][2 * 4 + k];
    }
    for (int s = 1; s <= 8; s++) {
      const int j0 = 2 * (s - 1);
      if (t >= j0) {
        At[0][s][t][0] = MN[t - j0][0]; At[0][s][t][1] = MN[t - j0][1];
        At[1][s][t][0] = MN[t - j0][4]; At[1][s][t][1] = MN[t - j0][5];
      }
      if (t >= j0 + 1) {
        At[0][s][t][2] = MN[t - j0 - 1][0]; At[0][s][t][3] = MN[t - j0 - 1][1];
        At[1][s][t][2] = MN[t - j0 - 1][4]; At[1][s][t][3] = MN[t - j0 - 1][5];
      }
    }
  }
  for (int r = 0; r < 4; r++) {
    for (int k = 0; k < 4; k++) At[2][0][r][k] = Mp[16][r * 4 + k];
    for (int s = 1; s <= 8; s++) {
      const int j0 = 2 * (s - 1);
      At[2][s][r][0] = MN[15 - j0][r * 2 + 0];
      At[2][s][r][1] = MN[15 - j0][r * 2 + 1];
      At[2][s][r][2] = MN[14 - j0][r * 2 + 0];
      At[2][s][r][3] = MN[14 - j0][r * 2 + 1];
    }
  }
  // swizzle into per-lane A layout: lane l holds row l%16; VGPR v holds K = (l<16?0:2)+v
  for (int out = 0; out < 3; out++)
    for (int s = 0; s < 9; s++)
      for (int l = 0; l < 32; l++)
        for (int v = 0; v < 2; v++) {
          const int row = l & 15;
          const int kc = ((l < 16) ? 0 : 2) + v;
          ws[WS_A + ((out * 9 + s) * 2 + v) * 32 + l] = At[out][s][row][kc];
        }
}

// ---------------- shared WMMA helpers ----------------
#define WMMA4(acc, a, b) \
  acc = __builtin_amdgcn_wmma_f32_16x16x4_f32(false, (a), false, (b), (short)0, (acc), false, false)

#define V8F_ZERO {0.f, 0.f, 0.f, 0.f, 0.f, 0.f, 0.f, 0.f}

// ---------------- K1 body: per-chunk aggregate r_c (carry-only WMMA) ----------------
template <bool GUARD>
__device__ __forceinline__ v8f k1_body(const float* __restrict__ u, const v2f* ac,
                                       int base, int joff, int lane) {
  v2f b0; b0.x = 0.f; b0.y = 0.f;  // x_base = 0 for the aggregate pass
  v8f carry = V8F_ZERO;
#pragma unroll
  for (int blk = 0; blk < NBLK; blk++) {
    v2f bs[8];
#pragma unroll
    for (int s = 0; s < 8; s++) {
      const int n = base + blk * 16 + 2 * s + joff;
      if (GUARD) {
        if (n < T_STEPS) bs[s] = *(const v2f*)(u + 2 * n);
        else { bs[s].x = 0.f; bs[s].y = 0.f; }
      } else {
        bs[s] = *(const v2f*)(u + 2 * n);
      }
    }
    v8f acc = V8F_ZERO;
    WMMA4(acc, ac[0], b0);
#pragma unroll
    for (int s = 0; s < 8; s++) WMMA4(acc, ac[s + 1], bs[s]);
    carry = acc;
    // rebuild B-slice-0 layout for next block: VGPR0 = state0|state2, VGPR1 = state1|state3
    const float sh2 = __shfl(carry[2], lane & 15, 32);
    const float sh3 = __shfl(carry[3], lane & 15, 32);
    b0.x = (lane < 16) ? carry[0] : sh2;
    b0.y = (lane < 16) ? carry[1] : sh3;
  }
  return carry;
}

__global__ void __launch_bounds__(256) k1_pass1(const float* __restrict__ u,
                                                float* __restrict__ ws) {
  const int lane = threadIdx.x & 31;
  const int wave = (blockIdx.x * 256 + threadIdx.x) >> 5;
  const int chunk = wave * 16 + (lane & 15);

  // waves whose 16 chunks are entirely past T: aggregates are zero, no loads needed
  if (wave * (16 * CLEN) >= T_STEPS) {
    if (lane < 16) {
      float* r = ws + WS_R + chunk * 4;
      r[0] = 0.f; r[1] = 0.f; r[2] = 0.f; r[3] = 0.f;
    }
    return;
  }

  const float* At = ws + WS_A;
  v2f ac[9];
#pragma unroll
  for (int s = 0; s < 9; s++) {
    ac[s].x = At[((2 * 9 + s) * 2 + 0) * 32 + lane];
    ac[s].y = At[((2 * 9 + s) * 2 + 1) * 32 + lane];
  }

  const int base = chunk * CLEN;
  const int joff = (lane >= 16) ? 1 : 0;
  v8f carry;
  if ((wave + 1) * (16 * CLEN) <= T_STEPS)      // wave-uniform: fully in-range hot path
    carry = k1_body<false>(u, ac, base, joff, lane);
  else                                          // single boundary wave
    carry = k1_body<true>(u, ac, base, joff, lane);

  if (lane < 16) {
    float* r = ws + WS_R + chunk * 4;
    r[0] = carry[0]; r[1] = carry[1]; r[2] = carry[2]; r[3] = carry[3];
  }
}

// ---------------- K2: matrix-weighted Kogge-Stone scan over chunk aggregates ----------------
__global__ void __launch_bounds__(1024) k2_scan(const float* __restrict__ x0,
                                                float* __restrict__ ws) {
  __shared__ float sv[1024 * 4];
  const int t = threadIdx.x;
  float ML[16];
#pragma unroll
  for (int i = 0; i < 16; i++) ML[i] = ws[WS_ML + i];
  const float* r = ws + WS_R;

  float v[4];
  if (t == 0) { v[0] = x0[0]; v[1] = x0[1]; v[2] = x0[2]; v[3] = x0[3]; }
  else { v[0] = v[1] = v[2] = v[3] = 0.f; }

  // local aggregate over this thread's 32 chunks
  for (int j = 0; j < 32; j++) {
    const float* rp = r + (t * 32 + j) * 4;
    float nv[4];
#pragma unroll
    for (int i = 0; i < 4; i++)
      nv[i] = ML[i * 4 + 0] * v[0] + ML[i * 4 + 1] * v[1] +
              ML[i * 4 + 2] * v[2] + ML[i * 4 + 3] * v[3] + rp[i];
    v[0] = nv[0]; v[1] = nv[1]; v[2] = nv[2]; v[3] = nv[3];
  }
#pragma unroll
  for (int i = 0; i < 4; i++) sv[t * 4 + i] = v[i];
  __syncthreads();

  for (int k = 0; k < 10; k++) {
    const int d = 1 << k;
    float w[4] = {0.f, 0.f, 0.f, 0.f};
    if (t >= d)
      for (int i = 0; i < 4; i++) w[i] = sv[(t - d) * 4 + i];
    __syncthreads();
    if (t >= d) {
      const float* P = ws + WS_P + k * 16;
#pragma unroll
      for (int i = 0; i < 4; i++)
        v[i] += P[i * 4 + 0] * w[0] + P[i * 4 + 1] * w[1] +
                P[i * 4 + 2] * w[2] + P[i * 4 + 3] * w[3];
      for (int i = 0; i < 4; i++) sv[t * 4 + i] = v[i];
    }
    __syncthreads();
  }

  // exclusive prefix = start state of this thread's first chunk; replay writes all starts
  float x[4];
  if (t == 0) { x[0] = x0[0]; x[1] = x0[1]; x[2] = x0[2]; x[3] = x0[3]; }
  else
    for (int i = 0; i < 4; i++) x[i] = sv[(t - 1) * 4 + i];

  float* xs = ws + WS_XS;
  for (int j = 0; j < 32; j++) {
    const int idx = t * 32 + j;
#pragma unroll
    for (int i = 0; i < 4; i++) xs[idx * 4 + i] = x[i];
    const float* rp = r + idx * 4;
    float nx[4];
#pragma unroll
    for (int i = 0; i < 4; i++)
      nx[i] = ML[i * 4 + 0] * x[0] + ML[i * 4 + 1] * x[1] +
              ML[i * 4 + 2] * x[2] + ML[i * 4 + 3] * x[3] + rp[i];
    x[0] = nx[0]; x[1] = nx[1]; x[2] = nx[2]; x[3] = nx[3];
  }
}

// ---------------- K3 body: replay with outputs (z1,z2 + carry = 27 WMMAs/block) ----------------
template <bool GUARD>
__device__ __forceinline__ void k3_body(const float* __restrict__ u, float* __restrict__ out,
                                        const v2f* az1, const v2f* az2, const v2f* ac,
                                        v2f b0, int base, int joff, int toff, int lane) {
#pragma unroll
  for (int blk = 0; blk < NBLK; blk++) {
    v2f bs[8];
#pragma unroll
    for (int s = 0; s < 8; s++) {
      const int n = base + blk * 16 + 2 * s + joff;
      if (GUARD) {
        if (n < T_STEPS) bs[s] = *(const v2f*)(u + 2 * n);
        else { bs[s].x = 0.f; bs[s].y = 0.f; }
      } else {
        bs[s] = *(const v2f*)(u + 2 * n);
      }
    }
    v8f a1 = V8F_ZERO, a2 = V8F_ZERO, cr = V8F_ZERO;
    WMMA4(a1, az1[0], b0);
    WMMA4(a2, az2[0], b0);
    WMMA4(cr, ac[0],  b0);
#pragma unroll
    for (int s = 0; s < 8; s++) {
      WMMA4(a1, az1[s + 1], bs[s]);
      WMMA4(a2, az2[s + 1], bs[s]);
      WMMA4(cr, ac[s + 1],  bs[s]);
    }
    // D rows = timesteps, columns = chunks: VGPR v -> t = v + (lane<16?0:8)
#pragma unroll
    for (int v = 0; v < 8; v++) {
      const int n = base + blk * 16 + v + toff;
      if (GUARD) {
        if (n < T_STEPS) {
          v2f o; o.x = a1[v]; o.y = a2[v];
          *(v2f*)(out + 2 * n) = o;
        }
      } else {
        v2f o; o.x = a1[v]; o.y = a2[v];
        *(v2f*)(out + 2 * n) = o;
      }
    }
    // carry -> next block's B-slice 0
    const float sh2 = __shfl(cr[2], lane & 15, 32);
    const float sh3 = __shfl(cr[3], lane & 15, 32);
    b0.x = (lane < 16) ? cr[0] : sh2;
    b0.y = (lane < 16) ? cr[1] : sh3;
  }
}

__global__ void __launch_bounds__(256) k3_pass3(const float* __restrict__ u,
                                                const float* __restrict__ ws,
                                                float* __restrict__ out) {
  const int lane = threadIdx.x & 31;
  const int wave = (blockIdx.x * 256 + threadIdx.x) >> 5;
  const int chunk = wave * 16 + (lane & 15);

  if (wave * (16 * CLEN) >= T_STEPS) return;  // fully past T: nothing to store

  const float* At = ws + WS_A;
  v2f az1[9], az2[9], ac[9];
#pragma unroll
  for (int s = 0; s < 9; s++) {
    az1[s].x = At[((0 * 9 + s) * 2 + 0) * 32 + lane];
    az1[s].y = At[((0 * 9 + s) * 2 + 1) * 32 + lane];
    az2[s].x = At[((1 * 9 + s) * 2 + 0) * 32 + lane];
    az2[s].y = At[((1 * 9 + s) * 2 + 1) * 32 + lane];
    ac[s].x  = At[((2 * 9 + s) * 2 + 0) * 32 + lane];
    ac[s].y  = At[((2 * 9 + s) * 2 + 1) * 32 + lane];
  }

  // initial x_base in B-slice-0 layout: lanes<16 -> states{0,1}, lanes>=16 -> states{2,3}
  const float* xs = ws + WS_XS + chunk * 4 + ((lane < 16) ? 0 : 2);
  v2f b0 = *(const v2f*)xs;

  const int base = chunk * CLEN;
  const int joff = (lane >= 16) ? 1 : 0;
  const int toff = (lane < 16) ? 0 : 8;

  if ((wave + 1) * (16 * CLEN) <= T_STEPS)      // wave-uniform: fully in-range hot path
    k3_body<false>(u, out, az1, az2, ac, b0, base, joff, toff, lane);
  else                                          // single boundary wave
    k3_body<true>(u, out, az1, az2, ac, b0, base, joff, toff, lane);
}

// ---------------- launch ----------------
extern "C" void kernel_launch(void* const* d_in, const int* in_sizes, int n_in,
                              void* d_out, int out_size, void* d_ws, size_t ws_size,
                              hipStream_t stream) {
  (void)in_sizes; (void)n_in; (void)out_size; (void)ws_size;
  const float* u  = (const float*)d_in[0];
  const float* x0 = (const float*)d_in[1];
  const float* c  = (const float*)d_in[2];
  const float* m  = (const float*)d_in[3];
  const float* kk = (const float*)d_in[4];
  const float* dt = (const float*)d_in[5];
  float* out = (float*)d_out;
  float* ws  = (float*)d_ws;

  k0_setup<<<1, 1, 0, stream>>>(c, m, kk, dt, ws);
  k1_pass1<<<CHUNKS / 128, 256, 0, stream>>>(u, ws);        // 2048 waves x 16 chunks
  k2_scan<<<1, 1024, 0, stream>>>(x0, ws);
  k3_pass3<<<CHUNKS / 128, 256, 0, stream>>>(u, ws, out);
}